// FineModule_82695300317910
// MI455X (gfx1250) — compile-verified
//
#include <hip/hip_runtime.h>
#include <hip/hip_bf16.h>

typedef float v2f __attribute__((ext_vector_type(2)));
typedef float v8f __attribute__((ext_vector_type(8)));

#define WW25 25
#define CCH 128
#define KB 2            // keypoints per workgroup
#define SIMS 33         // padded column stride for sim_T (col 25 is a zero column)

// ---- kernel 0: permute w1 (O,I,3,3) -> W9[t][o][i] so A-fragments are contiguous ----
__global__ void permute_w1(const float* __restrict__ w1, float* __restrict__ W9) {
    int idx = blockIdx.x * blockDim.x + threadIdx.x;    // over 128*128*9
    if (idx >= CCH * CCH * 9) return;
    int t = idx % 9;
    int i = (idx / 9) % CCH;
    int o = idx / (9 * CCH);
    W9[((size_t)t * CCH + o) * CCH + i] = w1[idx];
}

// ---- main fused kernel: att -> sim -> conv3x3 (WMMA f32) -> GN -> ReLU -> 1x1 -> softmax -> moments ----
__launch_bounds__(256, 2)
__global__ void fine_kernel(const float* __restrict__ kptsfeat1,
                            const float* __restrict__ feat,       // kptsfeat0_from1
                            const float* __restrict__ center,
                            const float* __restrict__ W9,
                            const float* __restrict__ b1,
                            const float* __restrict__ gnw,
                            const float* __restrict__ gnb,
                            const float* __restrict__ w2,
                            float* __restrict__ out, int Ktot)
{
    __shared__ float simT[KB][CCH * SIMS];   // sim transposed: [channel][position], col 25 == 0
    __shared__ float picked[KB][CCH];
    __shared__ float hm[KB][32];             // 1x1-conv logits accumulator
    __shared__ float sB1[CCH], sGW[CCH], sGB[CCH], sW2[CCH];

    const int t    = threadIdx.x;
    const int lane = t & 31;
    const int wave = t >> 5;
    const int kbase = blockIdx.x * KB;

    // ---------------- stage A: params + picked + zero pads ----------------
    if (t < CCH) { sB1[t] = b1[t]; sGW[t] = gnw[t]; sGB[t] = gnb[t]; sW2[t] = w2[t]; }
    for (int idx = t; idx < KB * CCH; idx += 256) {
        int q = idx >> 7, c = idx & 127;
        picked[q][c] = kptsfeat1[(size_t)(kbase + q) * WW25 * CCH + (WW25 / 2) * CCH + c];
        simT[q][c * SIMS + 25] = 0.0f;       // zero column for out-of-bounds im2col taps
    }
    if (t < KB * 32) hm[t >> 5][t & 31] = 0.0f;
    __syncthreads();

    // attention dot + sim, one (q, row) task per wave at a time
    for (int task = wave; task < KB * WW25; task += 8) {
        int q = task / WW25, w = task % WW25;
        const float* frow = feat + ((size_t)(kbase + q) * WW25 + w) * CCH;
        float4 v  = ((const float4*)frow)[lane];
        float4 pk = ((const float4*)&picked[q][0])[lane];
        float s = v.x * pk.x + v.y * pk.y + v.z * pk.z + v.w * pk.w;
        #pragma unroll
        for (int m = 16; m >= 1; m >>= 1) s += __shfl_xor(s, m, 32);
        int c0 = lane * 4;
        simT[q][(c0 + 0) * SIMS + w] = v.x * s;
        simT[q][(c0 + 1) * SIMS + w] = v.y * s;
        simT[q][(c0 + 2) * SIMS + w] = v.z * s;
        simT[q][(c0 + 3) * SIMS + w] = v.w * s;
    }
    __syncthreads();

    // ---------------- stage B: 3x3 conv as 9 tap-GEMMs with f32 WMMA ----------------
    v8f acc[KB * 2] = {};                        // [q][n-tile] accumulators (16x16 f32 each)
    const int half = lane >> 4;                  // K-pair / row-half selector
    const int ncol = lane & 15;                  // N column (C/D: N = lane%16)
    const int mrow = (wave << 4) + ncol;         // A row = output channel

    for (int tap = 0; tap < 9; ++tap) {
        const int dh = tap / 3 - 1, dw = tap % 3 - 1;
        int col0, col1;
        {
            int p = ncol;      int h = p / 5 + dh, w_ = p % 5 + dw;
            col0 = (h >= 0 && h < 5 && w_ >= 0 && w_ < 5) ? (h * 5 + w_) : 25;
            p = 16 + ncol;     h = p / 5 + dh;    w_ = p % 5 + dw;
            col1 = (h >= 0 && h < 5 && w_ >= 0 && w_ < 5) ? (h * 5 + w_) : 25;
        }
        const float* At = W9 + ((size_t)tap * CCH + mrow) * CCH;
        for (int ks = 0; ks < 32; ++ks) {
            const int i0 = ks * 4 + half * 2;    // this lane's K pair (A 16x4 layout)
            v2f a = *(const v2f*)(At + i0);
            #pragma unroll
            for (int q = 0; q < KB; ++q) {
                const float* S = &simT[q][0];
                v2f b0; b0.x = S[i0 * SIMS + col0]; b0.y = S[(i0 + 1) * SIMS + col0];
                v2f b1f; b1f.x = S[i0 * SIMS + col1]; b1f.y = S[(i0 + 1) * SIMS + col1];
                acc[q * 2 + 0] = __builtin_amdgcn_wmma_f32_16x16x4_f32(
                    false, a, false, b0, (short)0, acc[q * 2 + 0], false, false);
                acc[q * 2 + 1] = __builtin_amdgcn_wmma_f32_16x16x4_f32(
                    false, a, false, b1f, (short)0, acc[q * 2 + 1], false, false);
            }
        }
    }

    // ---------------- stage C: bias + GroupNorm + ReLU + 1x1 conv reduce ----------------
    const int chbase = wave * 16 + half * 8;     // C/D layout: VGPR j -> M = j + 8*half
    const bool p2ok = (16 + ncol) < 25;
    for (int q = 0; q < KB; ++q) {
        v8f a0 = acc[q * 2 + 0], a1 = acc[q * 2 + 1];
        #pragma unroll
        for (int j = 0; j < 8; ++j) { float bb = sB1[chbase + j]; a0[j] += bb; a1[j] += bb; }
        // group sums: group A = channels chbase+0..3, group B = chbase+4..7 (complete within this lane-half)
        float sA = 0.f, qA = 0.f, sB = 0.f, qB = 0.f;
        #pragma unroll
        for (int j = 0; j < 4; ++j) {
            float v = a0[j]; sA += v; qA += v * v;
            if (p2ok) { v = a1[j]; sA += v; qA += v * v; }
        }
        #pragma unroll
        for (int j = 4; j < 8; ++j) {
            float v = a0[j]; sB += v; qB += v * v;
            if (p2ok) { v = a1[j]; sB += v; qB += v * v; }
        }
        #pragma unroll
        for (int m = 1; m < 16; m <<= 1) {
            sA += __shfl_xor(sA, m, 32); qA += __shfl_xor(qA, m, 32);
            sB += __shfl_xor(sB, m, 32); qB += __shfl_xor(qB, m, 32);
        }
        const float inv100 = 1.0f / 100.0f;
        float muA = sA * inv100, ivA = rsqrtf(qA * inv100 - muA * muA + 1e-5f);
        float muB = sB * inv100, ivB = rsqrtf(qB * inv100 - muB * muB + 1e-5f);
        float part0 = 0.f, part1 = 0.f;
        #pragma unroll
        for (int j = 0; j < 8; ++j) {
            float mu = (j < 4) ? muA : muB, iv = (j < 4) ? ivA : ivB;
            int ch = chbase + j;
            float g = sGW[ch], bt = sGB[ch], wv = sW2[ch];
            part0 += fmaxf((a0[j] - mu) * iv * g + bt, 0.f) * wv;
            part1 += fmaxf((a1[j] - mu) * iv * g + bt, 0.f) * wv;
        }
        part0 += __shfl_xor(part0, 16, 32);      // fold in the other 8 channels (same p)
        part1 += __shfl_xor(part1, 16, 32);
        if (lane < 16) {
            atomicAdd(&hm[q][lane], part0);
            if (16 + lane < 25) atomicAdd(&hm[q][16 + lane], part1);
        }
    }
    __syncthreads();

    // ---------------- stage D: softmax + soft-argmax moments (b2 cancels in softmax) ----------------
    if (wave < KB) {
        const int q = wave;
        const size_t k = (size_t)kbase + q;
        float v = (lane < WW25) ? hm[q][lane] : -1e30f;
        float mx = v;
        #pragma unroll
        for (int m = 1; m < 32; m <<= 1) mx = fmaxf(mx, __shfl_xor(mx, m, 32));
        float e = (lane < WW25) ? expf(v - mx) : 0.f;
        float se = e;
        #pragma unroll
        for (int m = 1; m < 32; m <<= 1) se += __shfl_xor(se, m, 32);
        float prob = e / se;
        float xx = -1.0f + 0.5f * (float)(lane % 5);
        float xy = -1.0f + 0.5f * (float)(lane / 5);
        float px = prob * xx, py = prob * xy, sxx = prob * xx * xx, syy = prob * xy * xy;
        #pragma unroll
        for (int m = 1; m < 32; m <<= 1) {
            px += __shfl_xor(px, m, 32);  py += __shfl_xor(py, m, 32);
            sxx += __shfl_xor(sxx, m, 32); syy += __shfl_xor(syy, m, 32);
        }
        if (lane == 0) {
            float varx = sxx - px * px, vary = syy - py * py;
            float stdv = sqrtf(fmaxf(varx, 1e-10f)) + sqrtf(fmaxf(vary, 1e-10f));
            float cx = center[k * 2 + 0], cy = center[k * 2 + 1];
            out[2 * k + 0] = px;  out[2 * k + 1] = py;
            out[2 * (size_t)Ktot + 2 * k + 0] = cx + px * 2.0f;
            out[2 * (size_t)Ktot + 2 * k + 1] = cy + py * 2.0f;
            out[4 * (size_t)Ktot + k] = stdv;
        }
    }
}

extern "C" void kernel_launch(void* const* d_in, const int* in_sizes, int n_in,
                              void* d_out, int out_size, void* d_ws, size_t ws_size,
                              hipStream_t stream) {
    const float* kptsfeat1 = (const float*)d_in[0];
    const float* feat      = (const float*)d_in[1];
    const float* center    = (const float*)d_in[2];
    const float* w1        = (const float*)d_in[3];
    const float* b1        = (const float*)d_in[4];
    const float* gnw       = (const float*)d_in[5];
    const float* gnb       = (const float*)d_in[6];
    const float* w2        = (const float*)d_in[7];
    (void)n_in; (void)ws_size; (void)out_size;

    float* W9 = (float*)d_ws;                         // 9*128*128 f32 = 576 KB scratch
    const int Ktot = in_sizes[2] / 2;                 // patch0_center_coord is (K,2)

    permute_w1<<<(CCH * CCH * 9 + 255) / 256, 256, 0, stream>>>(w1, W9);
    fine_kernel<<<Ktot / KB, 256, 0, stream>>>(kptsfeat1, feat, center, W9,
                                               b1, gnw, gnb, w2,
                                               (float*)d_out, Ktot);
}